// TripletLossWithHardMining_64149631533276
// MI455X (gfx1250) — compile-verified
//
#include <hip/hip_runtime.h>
#include <stdint.h>

#define B_DIM 1024
#define N_DIM 1024
#define D_DIM 64
#define MARGIN 0.5f
#define EPSV 1e-6f

#define TILE_N 64       // columns staged per LDS tile
#define THREADS 128     // 4 waves of 32
#define WAVES 4
#define NEG_INF -3.0e38f

typedef __attribute__((ext_vector_type(2))) float v2f;
typedef __attribute__((ext_vector_type(8))) float v8f;

// Low 32 bits of a flat shared-aperture address are the LDS byte address.
__device__ __forceinline__ uint32_t lds_byte_addr(const void* p) {
    return (uint32_t)(uintptr_t)p;
}

// CDNA5 async global -> LDS DMA (ASYNCcnt-tracked), B128 per lane.
__device__ __forceinline__ void async_b128(uint32_t lds_addr, const void* gaddr) {
    asm volatile("global_load_async_to_lds_b128 %0, %1, off"
                 :: "v"(lds_addr), "v"(gaddr)
                 : "memory");
}

__device__ __forceinline__ void wait_async0() {
    asm volatile("s_wait_asynccnt 0" ::: "memory");
}

__device__ __forceinline__ float sel8(v8f c, int p) {
    float r = c[0];
    r = (p == 1) ? c[1] : r;
    r = (p == 2) ? c[2] : r;
    r = (p == 3) ? c[3] : r;
    r = (p == 4) ? c[4] : r;
    r = (p == 5) ? c[5] : r;
    r = (p == 6) ? c[6] : r;
    r = (p == 7) ? c[7] : r;
    return r;
}

__global__ void tl_init(float* loss_out) {
    if (threadIdx.x == 0) loss_out[0] = 0.0f;
}

// One block per batch row b. Computes dist_pos/dist_neg per column via WMMA
// Gram diagonals, reduces loss + hard-negative argmax for this b.
__global__ void __launch_bounds__(THREADS)
tl_phase1(const float* __restrict__ anchor,
          const float* __restrict__ positive,
          const float* __restrict__ negative,
          float* __restrict__ loss_out,
          float* __restrict__ hasvalid_out,
          int* __restrict__ selcol) {
    __shared__ __align__(16) float ldsA[TILE_N * D_DIM];
    __shared__ __align__(16) float ldsP[TILE_N * D_DIM];
    __shared__ __align__(16) float ldsN[TILE_N * D_DIM];
    __shared__ float s_loss[WAVES];
    __shared__ float s_best[WAVES];
    __shared__ int   s_idx[WAVES];

    const int b     = blockIdx.x;
    const int tid   = threadIdx.x;
    const int lane  = tid & 31;
    const int wave  = tid >> 5;
    const int m     = lane & 15;   // row within 16-row WMMA group
    const int khalf = lane >> 4;   // K-pair selector for f32 16x16x4 A layout

    const size_t rowBase = (size_t)b * N_DIM * D_DIM;

    float lossAcc   = 0.0f;
    float bestScore = NEG_INF;
    int   bestIdx   = 0x7fffffff;

    for (int tile = 0; tile < N_DIM / TILE_N; ++tile) {
        // ---- async stage: contiguous 16KB per array -> LDS ----
        const size_t tBase = rowBase + (size_t)tile * TILE_N * D_DIM; // float idx
        const uint32_t la = lds_byte_addr(ldsA);
        const uint32_t lp = lds_byte_addr(ldsP);
        const uint32_t ln = lds_byte_addr(ldsN);
#pragma unroll
        for (int i = 0; i < (TILE_N * D_DIM / 4) / THREADS; ++i) {
            const int e = i * THREADS + tid;              // float4 index
            const uint32_t lo = (uint32_t)e * 16u;        // byte offset
            const size_t go = (size_t)e * 4;              // float offset
            async_b128(la + lo, (const void*)(anchor   + tBase + go));
            async_b128(lp + lo, (const void*)(positive + tBase + go));
            async_b128(ln + lo, (const void*)(negative + tBase + go));
        }
        wait_async0();
        __syncthreads();

        // ---- per-wave 16-row Gram via V_WMMA_F32_16X16X4_F32 ----
        const int rbase = wave * 16;
        const float* Ar = &ldsA[(rbase + m) * D_DIM];
        const float* Pr = &ldsP[(rbase + m) * D_DIM];
        const float* Nr = &ldsN[(rbase + m) * D_DIM];

        v8f cpos = {};
        v8f cneg = {};
#pragma unroll
        for (int kk = 0; kk < D_DIM / 4; ++kk) {
            const int kc = kk * 4 + khalf * 2;
            v2f a2 = *(const v2f*)(Ar + kc);
            v2f p2 = *(const v2f*)(Pr + kc);
            v2f n2 = *(const v2f*)(Nr + kc);
            v2f vp = a2 - p2 + EPSV;
            v2f vn = a2 - n2 + EPSV;
            // Gram: for this layout A-frag == B-frag of V^T, so D += V·V^T
            cpos = __builtin_amdgcn_wmma_f32_16x16x4_f32(
                false, vp, false, vp, (short)0, cpos, false, false);
            cneg = __builtin_amdgcn_wmma_f32_16x16x4_f32(
                false, vn, false, vn, (short)0, cneg, false, false);
        }

        // ---- diagonal extraction: lanes 0-7 hold diag N=0..7 (vgpr=lane),
        //      lanes 24-31 hold diag N=8..15 (vgpr=lane-24) ----
        const int p = (lane < 8) ? lane : ((lane >= 24) ? (lane - 24) : 0);
        float dpos_h = sel8(cpos, p);
        float dneg_h = sel8(cneg, p);
        const int src = (m < 8) ? m : (m + 16);
        float dp2 = __shfl(dpos_h, src, 32);
        float dn2 = __shfl(dneg_h, src, 32);

        float dp = sqrtf(dp2);
        float dn = sqrtf(dn2);
        float diff = MARGIN + dp - dn;
        float le = fmaxf(0.0f, diff);
        if (lane < 16) {
            lossAcc += le;
            float score = (diff > 0.0f) ? dn : NEG_INF;
            int n = tile * TILE_N + rbase + m;
            if (score > bestScore || (score == bestScore && n < bestIdx)) {
                bestScore = score;
                bestIdx = n;
            }
        }
        __syncthreads(); // protect LDS before next tile's async writes
    }

    // ---- wave reduction (lanes >=16 carry neutral values) ----
#pragma unroll
    for (int off = 16; off >= 1; off >>= 1) {
        lossAcc += __shfl_xor(lossAcc, off, 32);
        float os = __shfl_xor(bestScore, off, 32);
        int   oi = __shfl_xor(bestIdx, off, 32);
        if (os > bestScore || (os == bestScore && oi < bestIdx)) {
            bestScore = os;
            bestIdx = oi;
        }
    }
    if (lane == 0) {
        s_loss[wave] = lossAcc;
        s_best[wave] = bestScore;
        s_idx[wave]  = bestIdx;
    }
    __syncthreads();
    if (tid == 0) {
        float ls = 0.0f;
        float bs = NEG_INF;
        int   bi = 0x7fffffff;
#pragma unroll
        for (int w = 0; w < WAVES; ++w) {
            ls += s_loss[w];
            if (s_best[w] > bs || (s_best[w] == bs && s_idx[w] < bi)) {
                bs = s_best[w];
                bi = s_idx[w];
            }
        }
        atomicAdd(loss_out, ls * (1.0f / ((float)B_DIM * (float)N_DIM)));
        selcol[b] = (bi == 0x7fffffff) ? 0 : bi;   // jnp.argmax of all -inf -> 0
        hasvalid_out[b] = (bs > -1.0e38f) ? 1.0f : 0.0f;
    }
}

// hard_neg[b] = negative[sel_col[b]]  (copy one [N,D] = 256KB row per block).
// dst base (d_out+1) is only 4B aligned -> coalesced dword copy.
__global__ void __launch_bounds__(256)
tl_gather(const float* __restrict__ negative,
          const int* __restrict__ selcol,
          float* __restrict__ hard_neg) {
    const int b = blockIdx.x;
    const size_t src = (size_t)selcol[b] * (size_t)(N_DIM * D_DIM);
    const size_t dst = (size_t)b * (size_t)(N_DIM * D_DIM);
    const float* __restrict__ s = negative + src;
    float* __restrict__ d = hard_neg + dst;
#pragma unroll 8
    for (int i = threadIdx.x; i < N_DIM * D_DIM; i += 256) {
        d[i] = s[i];
    }
}

extern "C" void kernel_launch(void* const* d_in, const int* in_sizes, int n_in,
                              void* d_out, int out_size, void* d_ws, size_t ws_size,
                              hipStream_t stream) {
    const float* anchor   = (const float*)d_in[0];
    const float* positive = (const float*)d_in[1];
    const float* negative = (const float*)d_in[2];

    float* out      = (float*)d_out;
    float* loss_out = out;                                      // [1]
    float* hard_neg = out + 1;                                  // [B,N,D]
    float* hasvalid = out + 1 + (size_t)B_DIM * N_DIM * D_DIM;  // [B]
    int*   selcol   = (int*)d_ws;

    hipLaunchKernelGGL(tl_init, dim3(1), dim3(64), 0, stream, loss_out);
    hipLaunchKernelGGL(tl_phase1, dim3(B_DIM), dim3(THREADS), 0, stream,
                       anchor, positive, negative, loss_out, hasvalid, selcol);
    hipLaunchKernelGGL(tl_gather, dim3(B_DIM), dim3(256), 0, stream,
                       negative, selcol, hard_neg);
}